// FRNN_parallel_7344394076731
// MI455X (gfx1250) — compile-verified
//
#include <hip/hip_runtime.h>
#include <math.h>

// ---------------------------------------------------------------------------
// FRNN leaky recurrence on MI455X (gfx1250).
//   per step: wr = r @ W.T + b ; u = Lam*wr (+ (1-Lam)*x on visible) ; r=tanh(u)
// W (64MB fp32) is L2-resident; each step streams it once from L2.
// GEMM core: V_WMMA_F32_16X16X4_F32 (exact fp32, matches reference numerics).
// r-tile staging: Tensor Data Mover (tensor_load_to_lds) with hardware LDS
// padding + double-buffered LDS, overlapped with compute via TENSORcnt.
// ---------------------------------------------------------------------------

typedef __attribute__((ext_vector_type(2))) float v2f;
typedef __attribute__((ext_vector_type(4))) float v4f;
typedef __attribute__((ext_vector_type(8))) float v8f;
typedef __attribute__((ext_vector_type(4))) unsigned su4;
typedef __attribute__((ext_vector_type(8))) unsigned su8;

#define V_DIM 1024
#define F_DIM 4096
#define P_DIM 64
#define T_DIM 1024

#define KC        128          // K-chunk staged in LDS
#define LDS_PITCH 132          // 128 + 4 pad -> conflict-free ds_load_b64
#define NCHUNK    (F_DIM / KC) // 32
#define WAVES_PER_WG 8
#define NCOLS_PER_WG (16 * WAVES_PER_WG)    // 128 output columns per WG
#define NUM_WG (F_DIM / NCOLS_PER_WG)       // 32 WGs per step

#define USE_TDM 1              // 0 -> proven synchronous VGPR staging fallback

// D = A(16x4 f32) * B(4x16 f32) + C(16x16 f32)  -- V_WMMA_F32_16X16X4_F32
static __device__ inline v8f wmma_f32_16x16x4(v2f a, v2f b, v8f c) {
#if defined(__gfx1250__) && __has_builtin(__builtin_amdgcn_wmma_f32_16x16x4_f32)
  return __builtin_amdgcn_wmma_f32_16x16x4_f32(
      /*neg_a=*/false, a, /*neg_b=*/false, b,
      /*c_mod=*/(short)0, c, /*reuse_a=*/false, /*reuse_b=*/false);
#else
  c[0] = fmaf(a[0], b[0], c[0]);
  return c;
#endif
}

#if USE_TDM
// TDM: DMA r[0:64, kc:kc+128] (fp32, row stride 4096) into LDS at lds_byte,
// inserting 4 dwords of padding after every 128 dwords (-> 132-float pitch).
static __device__ inline void tdm_load_chunk(const float* gsrc,
                                             unsigned lds_byte) {
  const unsigned long long ga = (unsigned long long)(uintptr_t)gsrc;
  su4 g0;
  g0[0] = 1u;                                   // count=1 user descriptor
  g0[1] = lds_byte;                             // lds_addr (bytes)
  g0[2] = (unsigned)(ga & 0xffffffffu);         // global_addr[31:0]
  g0[3] = (unsigned)(ga >> 32) | 0x80000000u;   // global_addr[56:32] | type=2
  su8 g1;
  g1[0] = (2u << 16)        // data_size = 4B
        | (1u << 20)        // pad_enable
        | (6u << 22)        // pad_interval: every 128 dwords
        | (3u << 25);       // pad_amount: 4 dwords
  g1[1] = (4096u & 0xffffu) << 16;  // tensor_dim0[15:0]  = 4096
  g1[2] = (64u   & 0xffffu) << 16;  // tensor_dim1[15:0]  = 64
  g1[3] = (128u) << 16;             // tile_dim0 = 128
  g1[4] = 64u;                      // tile_dim1 = 64, tile_dim2 = 0
  g1[5] = 4096u;                    // tensor_dim0_stride[31:0] = 4096
  g1[6] = 0u;
  g1[7] = 0u;
  asm volatile("tensor_load_to_lds %0, %1, null, null"
               :: "s"(g0), "s"(g1)
               : "memory");
}
#endif

__global__ void frnn_zero(float* __restrict__ r) {
  r[blockIdx.x * blockDim.x + threadIdx.x] = 0.0f;
}

__global__ __launch_bounds__(256) void frnn_step(
    const float* __restrict__ X,     // [T, V]
    const float* __restrict__ W,     // [F, F] row-major
    const float* __restrict__ bias,  // [F]
    const float* __restrict__ Lam,   // [P, F]
    const float* __restrict__ r_in,  // [P, F]
    float* __restrict__ r_out,       // [P, F]
    float* __restrict__ out,         // [P, V]
    int t) {
  __shared__ float ldsA[2][P_DIM * LDS_PITCH];  // double-buffered r tile

  const int tid    = threadIdx.x;
  const int lane   = tid & 31;
  const int wave   = tid >> 5;
  const int laneN  = lane & 15;    // column within 16x16 tile
  const int laneHi = lane >> 4;    // half-wave select
  const int n0     = blockIdx.x * NCOLS_PER_WG + wave * 16;
  const int j      = n0 + laneN;   // this lane's output column

  // Accumulators: 4 M-tiles (M = 0..63) x 16 columns.
  v8f acc[4];
  acc[0] = (v8f){}; acc[1] = (v8f){}; acc[2] = (v8f){}; acc[3] = (v8f){};

  // B[k][n] = W[j*F + k]; per-lane this wave reads K = k0 + 2*laneHi + {0,1}.
  const float* __restrict__ wbase = W + (size_t)j * F_DIM + 2 * laneHi;

#if USE_TDM
  if (wave == 0)  // one DMA per chunk per WG; prime the pipeline with chunk 0
    tdm_load_chunk(r_in, (unsigned)(uintptr_t)&ldsA[0][0]);
#endif

  for (int ch = 0; ch < NCHUNK; ++ch) {
    const int kc = ch * KC;
#if USE_TDM
    if (wave == 0) {
      if (ch + 1 < NCHUNK) {
        tdm_load_chunk(r_in + (size_t)(kc + KC),
                       (unsigned)(uintptr_t)&ldsA[(ch + 1) & 1][0]);
        __builtin_amdgcn_s_wait_tensorcnt(1);  // in-order: chunk `ch` landed
      } else {
        __builtin_amdgcn_s_wait_tensorcnt(0);
      }
    }
    __syncthreads();                           // chunk `ch` visible to all
    const float* __restrict__ As = &ldsA[ch & 1][0];
#else
#pragma unroll
    for (int pass = 0; pass < 8; ++pass) {
      const int p   = pass * 8 + wave;
      const int kk4 = lane * 4;
      v4f v = *(const v4f*)(r_in + (size_t)p * F_DIM + kc + kk4);
      *(v4f*)(&ldsA[0][p * LDS_PITCH + kk4]) = v;
    }
    __syncthreads();
    const float* __restrict__ As = &ldsA[0][0];
#endif

    // K loop: 4 ds_load_b64 (A) + 1 global b64 (B) + 4 WMMA per 4-K step.
#pragma unroll 4
    for (int ks = 0; ks < KC; ks += 4) {
      const int kl = ks + 2 * laneHi;
      v2f b  = *(const v2f*)(wbase + kc + ks);
      v2f a0 = *(const v2f*)(&As[(laneN +  0) * LDS_PITCH + kl]);
      v2f a1 = *(const v2f*)(&As[(laneN + 16) * LDS_PITCH + kl]);
      v2f a2 = *(const v2f*)(&As[(laneN + 32) * LDS_PITCH + kl]);
      v2f a3 = *(const v2f*)(&As[(laneN + 48) * LDS_PITCH + kl]);
      acc[0] = wmma_f32_16x16x4(a0, b, acc[0]);
      acc[1] = wmma_f32_16x16x4(a1, b, acc[1]);
      acc[2] = wmma_f32_16x16x4(a2, b, acc[2]);
      acc[3] = wmma_f32_16x16x4(a3, b, acc[3]);
    }
    __syncthreads();                           // release this LDS buffer
  }

  // -- Epilogue: u = Lam*(wr + b) [+ (1-Lam)*x visible], r = tanh(u) --------
  const float bj  = bias[j];
  const bool  vis = (j < V_DIM);
  const float xj  = vis ? X[(size_t)t * V_DIM + j] : 0.0f;

#pragma unroll
  for (int mt = 0; mt < 4; ++mt) {
#pragma unroll
    for (int i = 0; i < 8; ++i) {
      const int p  = mt * 16 + i + 8 * laneHi;   // C/D layout: M = i + 8*hi
      const float wr  = acc[mt][i] + bj;
      const float lam = Lam[(size_t)p * F_DIM + j];
      float u = lam * wr;
      if (vis) u += (1.0f - lam) * xj;
      r_out[(size_t)p * F_DIM + j] = tanhf(u);
      if ((t == T_DIM - 1) && vis) out[(size_t)p * V_DIM + j] = u;
    }
  }
}

extern "C" void kernel_launch(void* const* d_in, const int* in_sizes, int n_in,
                              void* d_out, int out_size, void* d_ws,
                              size_t ws_size, hipStream_t stream) {
  (void)in_sizes; (void)n_in; (void)out_size; (void)ws_size;
  const float* X   = (const float*)d_in[0];   // [T, V]
  const float* W   = (const float*)d_in[1];   // [F, F]
  const float* b   = (const float*)d_in[2];   // [F]
  const float* Lam = (const float*)d_in[3];   // [P, F]
  float* out = (float*)d_out;

  // Ping-pong r buffers in workspace: 2 * P*F floats = 2 MB.
  float* rA = (float*)d_ws;
  float* rB = rA + (size_t)P_DIM * F_DIM;

  frnn_zero<<<(P_DIM * F_DIM) / 256, 256, 0, stream>>>(rA);

  for (int t = 0; t < T_DIM; ++t) {
    const float* rin = (t & 1) ? rB : rA;
    float*       rou = (t & 1) ? rA : rB;
    frnn_step<<<dim3(NUM_WG), dim3(256), 0, stream>>>(X, W, b, Lam, rin, rou,
                                                      out, t);
  }
}